// BaseNeighborlist_2156073583103
// MI455X (gfx1250) — compile-verified
//
#include <hip/hip_runtime.h>
#include <stdint.h>

// Problem constants (match reference)
#define CUTOFF_F    5.2f
#define MOLECULES   32
#define ATOMS       512
#define NATOMS      (MOLECULES * ATOMS)        // 16384 atoms
#define COORD_BYTES (NATOMS * 3 * 4)           // 196608 B = 192 KB, fits 320 KB LDS/WGP
#define NPAIRS      4194304u

#define BLOCKS      512
#define THREADS     512
#define FILL_ITERS  (COORD_BYTES / (THREADS * 16))   // = 24, exact (no remainder)

typedef int v4i_ __attribute__((ext_vector_type(4)));

// ---------------------------------------------------------------------------
// CDNA5 async global->LDS copy (ASYNCcnt-tracked), builtin with asm fallback.
// Toolchain signature (probe-verified): (int4 AS1*, int4 AS3*, imm, imm cpol).
// ---------------------------------------------------------------------------
__device__ __forceinline__ void async_load_b128_to_lds(const void* g, void* lds) {
#if defined(__HIP_DEVICE_COMPILE__)
#if __has_builtin(__builtin_amdgcn_global_load_async_to_lds_b128)
  __builtin_amdgcn_global_load_async_to_lds_b128(
      (__attribute__((address_space(1))) v4i_*)g,
      (__attribute__((address_space(3))) v4i_*)lds,
      /*imm offset=*/0, /*cpol=*/0);
#else
  unsigned lds_off = (unsigned)(uintptr_t)lds;       // low 32 bits = LDS offset
  unsigned long long ga = (unsigned long long)(uintptr_t)g;
  asm volatile("global_load_async_to_lds_b128 %0, %1, off"
               :: "v"(lds_off), "v"(ga) : "memory");
#endif
#endif
}

__device__ __forceinline__ void wait_asynccnt0() {
#if defined(__HIP_DEVICE_COMPILE__)
#if __has_builtin(__builtin_amdgcn_s_wait_asynccnt)
  __builtin_amdgcn_s_wait_asynccnt(0);
#else
  asm volatile("s_wait_asynccnt 0" ::: "memory");
#endif
#endif
}

// ---------------------------------------------------------------------------
// Kernel: stage 192KB coord table in LDS (CDNA5 async copies), then screen
// pairs with LDS gathers + NT-hinted coalesced streaming I/O.
// ---------------------------------------------------------------------------
extern "C" __global__ void __launch_bounds__(THREADS, 1)
nbl_screen_kernel(const float* __restrict__ coords,   // (NATOMS, 3) flat
                  const int*   __restrict__ nl,       // (2, NPAIRS) int32
                  float*       __restrict__ out)      // [diff 3P | dist P | mask P]
{
  extern __shared__ __align__(16) char s_raw[];
  float* s_xyz = (float*)s_raw;

  const unsigned tid = threadIdx.x;

  // ---- Phase 1: async-fill the whole coordinate table into LDS ----
  // Exact compile-time trip count (24): straight-line issue, 24 B128 copies
  // in flight per lane on ASYNCcnt (limit 63), zero loop overhead.
  {
    const char* gsrc = (const char*)coords + tid * 16u;
    char*       ldst = s_raw + tid * 16u;
    #pragma unroll
    for (int k = 0; k < FILL_ITERS; ++k) {
      async_load_b128_to_lds(gsrc + k * (THREADS * 16), ldst + k * (THREADS * 16));
    }
  }
  wait_asynccnt0();   // this wave's async LDS writes are visible
  __syncthreads();    // all waves' fills complete

  // ---- Phase 2: grid-stride pair screening ----
  const int* __restrict__ row0 = nl;
  const int* __restrict__ row1 = nl + NPAIRS;
  float* __restrict__ out_diff = out;                       // 3*NPAIRS floats
  float* __restrict__ out_dist = out + 3u * NPAIRS;         // NPAIRS floats
  float* __restrict__ out_mask = out + 4u * NPAIRS;         // NPAIRS floats (bool as 0/1)

  const unsigned stride = gridDim.x * blockDim.x;
  for (unsigned p = blockIdx.x * blockDim.x + tid; p < NPAIRS; p += stride) {
    // Streaming, read-once: non-temporal loads keep L2 for the coord table
    const unsigned i = (unsigned)__builtin_nontemporal_load(row0 + p);
    const unsigned j = (unsigned)__builtin_nontemporal_load(row1 + p);

    // LDS gathers (random index -> served from 320KB LDS, not L2)
    const float x0 = s_xyz[3u * i + 0u];
    const float y0 = s_xyz[3u * i + 1u];
    const float z0 = s_xyz[3u * i + 2u];
    const float x1 = s_xyz[3u * j + 0u];
    const float y1 = s_xyz[3u * j + 1u];
    const float z1 = s_xyz[3u * j + 2u];

    const float dx = x0 - x1;
    const float dy = y0 - y1;
    const float dz = z0 - z1;
    const float d2 = dx * dx + dy * dy + dz * dz;
    const float dist = __builtin_sqrtf(d2);
    const bool  keep = (dist <= CUTOFF_F);

    // Write-once outputs: non-temporal stores (TH=NT), coalesced per wave
    __builtin_nontemporal_store(keep ? dx : 0.0f,   out_diff + 3u * p + 0u);
    __builtin_nontemporal_store(keep ? dy : 0.0f,   out_diff + 3u * p + 1u);
    __builtin_nontemporal_store(keep ? dz : 0.0f,   out_diff + 3u * p + 2u);
    __builtin_nontemporal_store(keep ? dist : 0.0f, out_dist + p);
    __builtin_nontemporal_store(keep ? 1.0f : 0.0f, out_mask + p);
  }
}

// ---------------------------------------------------------------------------
extern "C" void kernel_launch(void* const* d_in, const int* in_sizes, int n_in,
                              void* d_out, int out_size, void* d_ws, size_t ws_size,
                              hipStream_t stream) {
  const float* coords = (const float*)d_in[0];  // (32,512,3) float32
  const int*   nl     = (const int*)d_in[1];    // (2, 4194304) int
  float*       out    = (float*)d_out;          // [diff | dist | mask]

  nbl_screen_kernel<<<dim3(BLOCKS), dim3(THREADS), COORD_BYTES, stream>>>(coords, nl, out);
}